// LGCN_Encoder_7095285973816
// MI455X (gfx1250) — compile-verified
//
#include <hip/hip_runtime.h>

// ---------------------------------------------------------------------------
// LightGCN encoder, MI455X (gfx1250).
// Reference quirk: `ego` is never updated inside the layer loop, so
//   stacked       = [ego, A*ego, A*ego, A*ego]
//   path_stacked  = [ego, R*ego, R*ego, R*ego]
//   mean          = 0.25*ego + 0.75*(A*ego)
// => two SpMMs (NNZ=1e6, D=64) + replication. Memory-bound (~420 MB HBM
// traffic, ~18us floor @ 23.3 TB/s). WMMA inapplicable (sparse scatter);
// use async global->LDS edge staging (confirmed: global_load_async_to_lds_b32
// + s_wait_asynccnt in disasm), f32 global atomics on L2-resident
// accumulators, and non-temporal stores for the 345 MB streamed output.
// This revision removes all divergent control flow from the scatter hot loop
// (zero-val padding for the tail + cndmask pointer select, full unroll).
// ---------------------------------------------------------------------------

namespace {
constexpr int  kU   = 100000;
constexpr int  kI   = 50000;
constexpr int  kN   = 150000;   // kU + kI
constexpr int  kD   = 64;
constexpr long kNNZ = 1000000;
// d_out layout (floats): user_all[U*64] | item_all[I*64] | stacked[N*4*64] | path_stacked[N*4*64]
constexpr long kStackedOff = (long)kN * kD;                    //  9,600,000
constexpr long kPathOff    = kStackedOff + (long)kN * 4 * kD;  // 48,000,000
}

typedef float f4 __attribute__((ext_vector_type(4)));

#define GAS __attribute__((address_space(1)))
#define LAS __attribute__((address_space(3)))

#if defined(__HIP_DEVICE_COMPILE__) && __has_builtin(__builtin_amdgcn_global_load_async_to_lds_b32)
#define HAVE_ASYNC_LDS 1
#else
#define HAVE_ASYNC_LDS 0
#endif

// --- SpMM scatter. One block handles 256 edges: stage (row,col,val) into LDS
// via async global->LDS, then 16 lanes per edge gather a coalesced 256B row
// of the embedding table and scatter with f32 atomics.
// Tail blocks: staging index is clamped (duplicates last edge) and the
// duplicate slots get val=0, so the hot loop has NO branches at all.
// `dst` is pre-offset to slot 1 (node stride = 256 floats).
__global__ void __launch_bounds__(256)
lgcn_scatter(const int* __restrict__ rows,
             const int* __restrict__ cols,
             const float* __restrict__ vals,
             const float* __restrict__ ue,
             const float* __restrict__ ie,
             float* __restrict__ dst) {
  __shared__ int   lr[256];
  __shared__ int   lc[256];
  __shared__ float lv[256];
  const int  t    = threadIdx.x;
  const long base = (long)blockIdx.x * 256;
  long idx = base + t;
  if (idx >= kNNZ) idx = kNNZ - 1;   // clamp; duplicates neutralized below

#if HAVE_ASYNC_LDS
  __builtin_amdgcn_global_load_async_to_lds_b32(
      (GAS int*)(rows + idx), (LAS int*)(lr + t), 0, 0);
  __builtin_amdgcn_global_load_async_to_lds_b32(
      (GAS int*)(cols + idx), (LAS int*)(lc + t), 0, 0);
  __builtin_amdgcn_global_load_async_to_lds_b32(
      (GAS int*)(vals + idx), (LAS int*)(lv + t), 0, 0);
#if __has_builtin(__builtin_amdgcn_s_wait_asynccnt)
  __builtin_amdgcn_s_wait_asynccnt(0);
#else
  asm volatile("s_wait_asynccnt 0" ::: "memory");
#endif
  // Async data has landed; neutralize duplicated tail edges (ordered before
  // consumers by the barrier's DS-count wait).
  if (base + t >= kNNZ) lv[t] = 0.0f;
#else
  lr[t] = rows[idx];
  lc[t] = cols[idx];
  lv[t] = (base + t < kNNZ) ? vals[idx] : 0.0f;
#endif
  __syncthreads();

  const int sub = t >> 4;        // which edge within a group of 16
  const int c   = (t & 15) * 4;  // float4 chunk of the 64-dim row
#pragma unroll
  for (int i = 0; i < 16; ++i) {
    const int e = i * 16 + sub;
    const int   row = lr[e];
    const int   col = lc[e];
    const float v   = lv[e];
    const bool  isU = col < kU;
    const float* bp  = isU ? ue : ie;            // v_cndmask, no branch
    const long  ridx = isU ? (long)col : (long)col - kU;
    f4 x = *(const f4*)(bp + ridx * kD + c);
    float* d = dst + (long)row * 256 + c;
    atomicAdd(d + 0, v * x[0]);
    atomicAdd(d + 1, v * x[1]);
    atomicAdd(d + 2, v * x[2]);
    atomicAdd(d + 3, v * x[3]);
  }
}

// --- Zero the two in-place accumulator regions (slot 1 of each node's 4x64
// stack). Everything else in d_out is overwritten later.
__global__ void __launch_bounds__(256)
lgcn_zero(float* __restrict__ sa, float* __restrict__ sb) {
  long tid  = (long)blockIdx.x * blockDim.x + threadIdx.x;   // 2*N*16 threads
  long half = (long)kN * 16;
  float* base = (tid < half) ? sa : sb;
  long t = (tid < half) ? tid : tid - half;
  long n = t >> 4;
  int  c = (int)(t & 15) * 4;
  f4 z = {0.0f, 0.0f, 0.0f, 0.0f};
  *(f4*)(base + n * 256 + c) = z;
}

// --- Replicate slot1 -> slots 2,3; write slot0 = ego; write mean into
// contiguous user_all||item_all. All streaming writes are non-temporal.
__global__ void __launch_bounds__(256)
lgcn_finalize(const float* __restrict__ ue,
              const float* __restrict__ ie,
              float* __restrict__ out) {
  long tid = (long)blockIdx.x * blockDim.x + threadIdx.x;  // N*16 threads
  long n = tid >> 4;
  int  c = (int)(tid & 15) * 4;
  const float* esrc = (n < kU) ? (ue + n * kD) : (ie + (n - kU) * kD);
  f4 e = *(const f4*)(esrc + c);
  float* st = out + kStackedOff + n * 256 + c;
  float* ps = out + kPathOff    + n * 256 + c;
  f4 a = *(const f4*)(st + 64);   // A*ego (accumulated in place)
  f4 r = *(const f4*)(ps + 64);   // R*ego
  __builtin_nontemporal_store(e, (f4*)(st));
  __builtin_nontemporal_store(a, (f4*)(st + 128));
  __builtin_nontemporal_store(a, (f4*)(st + 192));
  __builtin_nontemporal_store(e, (f4*)(ps));
  __builtin_nontemporal_store(r, (f4*)(ps + 128));
  __builtin_nontemporal_store(r, (f4*)(ps + 192));
  f4 m = 0.25f * e + 0.75f * a;
  __builtin_nontemporal_store(m, (f4*)(out + n * kD + c));
}

extern "C" void kernel_launch(void* const* d_in, const int* in_sizes, int n_in,
                              void* d_out, int out_size, void* d_ws, size_t ws_size,
                              hipStream_t stream) {
  (void)in_sizes; (void)n_in; (void)out_size; (void)d_ws; (void)ws_size;
  const float* ue = (const float*)d_in[0];
  const float* ie = (const float*)d_in[1];
  const int*   ar = (const int*)d_in[2];
  const int*   ac = (const int*)d_in[3];
  const float* av = (const float*)d_in[4];
  const int*   rr = (const int*)d_in[5];
  const int*   rc = (const int*)d_in[6];
  const float* rv = (const float*)d_in[7];
  float* out     = (float*)d_out;
  float* stacked = out + kStackedOff;
  float* path    = out + kPathOff;

  const int threads = 256;
  const int zeroBlocks  = (int)((2L * kN * 16) / threads);        // 18750
  const int scatBlocks  = (int)((kNNZ + 255) / 256);              // 3907
  const int finBlocks   = (int)(((long)kN * 16) / threads);       // 9375

  lgcn_zero<<<zeroBlocks, threads, 0, stream>>>(stacked + 64, path + 64);
  lgcn_scatter<<<scatBlocks, threads, 0, stream>>>(ar, ac, av, ue, ie, stacked + 64);
  lgcn_scatter<<<scatBlocks, threads, 0, stream>>>(rr, rc, rv, ue, ie, path + 64);
  lgcn_finalize<<<finBlocks, threads, 0, stream>>>(ue, ie, out);
}